// MessageNet_5781025980521
// MI455X (gfx1250) — compile-verified
//
#include <hip/hip_runtime.h>
#include <math.h>

#define NN 100000      // nodes
#define NE 1000000     // edges
#define DF 64          // node feature dim
#define HF 128         // hidden dim

#define EW 16          // edges per wave (WMMA M)
#define NPW 16         // nodes per wave
#define WPB 4          // waves per block (128 threads, wave32)
#define ROWP 130       // padded row stride (2D + 2) for edge LDS tile
#define NROWP 196      // padded row stride (3D + 4) for node LDS tile

typedef float v2f __attribute__((ext_vector_type(2)));
typedef float v8f __attribute__((ext_vector_type(8)));

__device__ __forceinline__ v8f wmma_f32(v2f a, v2f b, v8f c) {
  // V_WMMA_F32_16X16X4_F32 : D = A(16x4) x B(4x16) + C, full fp32
  return __builtin_amdgcn_wmma_f32_16x16x4_f32(
      /*neg_a=*/false, a, /*neg_b=*/false, b,
      /*c_mod=*/(short)0, c, /*reuse_a=*/false, /*reuse_b=*/false);
}

// float atomic-max via sign-split int atomics (native global_atomic_max_i32 /
// min_u32). Correct for buffers initialized to -inf (0xFF800000).
__device__ __forceinline__ void atomicMaxFloat(float* addr, float v) {
  if (v >= 0.0f) atomicMax((int*)addr, __float_as_int(v));
  else           atomicMin((unsigned int*)addr, __float_as_uint(v));
}

__global__ void init_kernel(float* m_sum, unsigned int* m_max_bits, int n) {
  int i = blockIdx.x * blockDim.x + threadIdx.x;
  if (i < n) { m_sum[i] = 0.0f; m_max_bits[i] = 0xFF800000u; }  // -inf
}

// ---------------- Edge kernel: gate = sigmoid(MLP([xs|xe])); scatter e*xs ----
__global__ void __launch_bounds__(128)
edge_kernel(const float* __restrict__ x, const int* __restrict__ start,
            const int* __restrict__ end,
            const float* __restrict__ We1, const float* __restrict__ be1,
            const float* __restrict__ We2, const float* __restrict__ be2,
            float* __restrict__ m_sum, float* __restrict__ m_max)
{
  __shared__ float s_in[WPB][EW][ROWP];   // [edge][0:64)=xs, [64:128)=xe
  __shared__ float s_gate[WPB][EW];

  const int wave = threadIdx.x >> 5;
  const int lane = threadIdx.x & 31;
  const int half = lane >> 4;             // 0: xs lanes, 1: xe lanes
  const int tileBase = (blockIdx.x * WPB + wave) * EW;

  // ---- stage 16 edges: lanes 0-15 gather xs (float4 each), 16-31 gather xe
  const int c4 = (lane & 15) * 4;
  for (int i = 0; i < EW; ++i) {
    int e = tileBase + i; if (e >= NE) e = NE - 1;
    const int node = half ? end[e] : start[e];
    const float4 v = *(const float4*)(x + (size_t)node * DF + c4);
    *(float4*)&s_in[wave][i][half * DF + c4] = v;
  }
  __syncthreads();

  // ---- GEMM1: [16 edges x 128] @ We1[128 x 128], fp32 WMMA K=4 steps
  const int rowA = lane & 15;
  const int kk   = half * 2;
  const int colB = lane & 15;
  v8f zero = {};
  v8f acc[8];
  #pragma unroll
  for (int t = 0; t < 8; ++t) acc[t] = zero;

  for (int k0 = 0; k0 < 2 * DF; k0 += 4) {
    v2f a;
    a.x = s_in[wave][rowA][k0 + kk];
    a.y = s_in[wave][rowA][k0 + kk + 1];
    const float* B0 = We1 + (size_t)(k0 + kk) * HF + colB;   // L2-hot weights
    #pragma unroll
    for (int t = 0; t < 8; ++t) {
      v2f b;
      b.x = B0[t * 16];
      b.y = B0[HF + t * 16];
      acc[t] = wmma_f32(a, b, acc[t]);
    }
  }

  // ---- bias + ReLU + dot with We2 (H->1) folded into a lane reduction
  float part[8];
  #pragma unroll
  for (int j = 0; j < 8; ++j) part[j] = 0.0f;
  #pragma unroll
  for (int t = 0; t < 8; ++t) {
    const float bb = be1[t * 16 + colB];
    const float w2 = We2[t * 16 + colB];
    #pragma unroll
    for (int j = 0; j < 8; ++j) {
      float hv = fmaxf(acc[t][j] + bb, 0.0f);
      part[j] = fmaf(hv, w2, part[j]);
    }
  }
  // reduce across the 16 lanes of each half (rows j / j+8), wave32 shuffles
  #pragma unroll
  for (int j = 0; j < 8; ++j) {
    part[j] += __shfl_xor(part[j], 1, 32);
    part[j] += __shfl_xor(part[j], 2, 32);
    part[j] += __shfl_xor(part[j], 4, 32);
    part[j] += __shfl_xor(part[j], 8, 32);
  }
  {
    const float logit = part[lane & 7] + be2[0];
    const float g = 1.0f / (1.0f + expf(-logit));
    if ((lane & 15) < 8) s_gate[wave][half * 8 + (lane & 7)] = g;
  }
  __syncthreads();

  // ---- scatter gated messages: 2 dims per lane, native f32 atomics into L2
  const int d0 = lane * 2;
  for (int i = 0; i < EW; ++i) {
    const int e = tileBase + i;
    if (e >= NE) break;
    const float g = s_gate[wave][i];
    const size_t base = (size_t)end[e] * DF + d0;
    const float m0 = g * s_in[wave][i][d0];
    const float m1 = g * s_in[wave][i][d0 + 1];
    atomicAdd(&m_sum[base],     m0);
    atomicAdd(&m_sum[base + 1], m1);
    atomicMaxFloat(&m_max[base],     m0);
    atomicMaxFloat(&m_max[base + 1], m1);
  }
}

// ---------------- Node kernel: out = MLP([m_max|m_sum|x]) + x ---------------
__global__ void __launch_bounds__(128)
node_kernel(const float* __restrict__ x,
            const float* __restrict__ m_sum, const float* __restrict__ m_max,
            const float* __restrict__ Wn1, const float* __restrict__ bn1,
            const float* __restrict__ Wn2, const float* __restrict__ bn2,
            float* __restrict__ out)
{
  __shared__ float s_a[WPB][NPW][NROWP];  // cols [0:64)=m_max, [64:128)=m_sum, [128:192)=x

  const int wave = threadIdx.x >> 5;
  const int lane = threadIdx.x & 31;
  const int half = lane >> 4;
  const int tileBase = (blockIdx.x * WPB + wave) * NPW;
  const int d0 = lane * 2;

  // ---- stage [m_max' | m_sum | x] for 16 nodes (2 cols per lane per block)
  for (int i = 0; i < NPW; ++i) {
    int nd = tileBase + i; if (nd >= NN) nd = NN - 1;
    const size_t b = (size_t)nd * DF + d0;
    float mx0 = m_max[b], mx1 = m_max[b + 1];
    if (__float_as_uint(mx0) == 0xFF800000u) mx0 = 0.0f;  // empty segment
    if (__float_as_uint(mx1) == 0xFF800000u) mx1 = 0.0f;
    s_a[wave][i][d0]              = mx0;
    s_a[wave][i][d0 + 1]          = mx1;
    s_a[wave][i][DF + d0]         = m_sum[b];
    s_a[wave][i][DF + d0 + 1]     = m_sum[b + 1];
    s_a[wave][i][2 * DF + d0]     = x[b];
    s_a[wave][i][2 * DF + d0 + 1] = x[b + 1];
  }
  __syncthreads();

  const int rowA = lane & 15;
  const int kk   = half * 2;
  const int colB = lane & 15;
  v8f zero = {};

  // ---- GEMM1: [16 x 192] @ Wn1[192 x 128]
  v8f acc[8];
  #pragma unroll
  for (int t = 0; t < 8; ++t) acc[t] = zero;
  for (int k0 = 0; k0 < 3 * DF; k0 += 4) {
    v2f a;
    a.x = s_a[wave][rowA][k0 + kk];
    a.y = s_a[wave][rowA][k0 + kk + 1];
    const float* B0 = Wn1 + (size_t)(k0 + kk) * HF + colB;
    #pragma unroll
    for (int t = 0; t < 8; ++t) {
      v2f b;
      b.x = B0[t * 16];
      b.y = B0[HF + t * 16];
      acc[t] = wmma_f32(a, b, acc[t]);
    }
  }
  __syncthreads();

  // ---- ReLU(h) written back into freed LDS cols [0:128) (C-layout -> A-layout)
  #pragma unroll
  for (int t = 0; t < 8; ++t) {
    const float bb = bn1[t * 16 + colB];
    #pragma unroll
    for (int j = 0; j < 8; ++j) {
      const float hv = fmaxf(acc[t][j] + bb, 0.0f);
      s_a[wave][j + 8 * half][t * 16 + colB] = hv;
    }
  }
  __syncthreads();

  // ---- GEMM2: [16 x 128] @ Wn2[128 x 64]
  v8f acc2[4];
  #pragma unroll
  for (int t = 0; t < 4; ++t) acc2[t] = zero;
  for (int k0 = 0; k0 < HF; k0 += 4) {
    v2f a;
    a.x = s_a[wave][rowA][k0 + kk];
    a.y = s_a[wave][rowA][k0 + kk + 1];
    const float* B0 = Wn2 + (size_t)(k0 + kk) * DF + colB;
    #pragma unroll
    for (int t = 0; t < 4; ++t) {
      v2f b;
      b.x = B0[t * 16];
      b.y = B0[DF + t * 16];
      acc2[t] = wmma_f32(a, b, acc2[t]);
    }
  }

  // ---- epilogue: + bn2 + residual x (kept in LDS cols [128:192))
  #pragma unroll
  for (int t = 0; t < 4; ++t) {
    const int c = t * 16 + colB;
    const float bb = bn2[c];
    #pragma unroll
    for (int j = 0; j < 8; ++j) {
      const int m = j + 8 * half;
      const int nd = tileBase + m;
      if (nd < NN)
        out[(size_t)nd * DF + c] = acc2[t][j] + bb + s_a[wave][m][2 * DF + c];
    }
  }
}

extern "C" void kernel_launch(void* const* d_in, const int* in_sizes, int n_in,
                              void* d_out, int out_size, void* d_ws, size_t ws_size,
                              hipStream_t stream) {
  const float* x   = (const float*)d_in[0];
  const int* start = (const int*)d_in[1];
  const int* end   = (const int*)d_in[2];
  const float* We1 = (const float*)d_in[3];
  const float* be1 = (const float*)d_in[4];
  const float* We2 = (const float*)d_in[5];
  const float* be2 = (const float*)d_in[6];
  const float* Wn1 = (const float*)d_in[7];
  const float* bn1 = (const float*)d_in[8];
  const float* Wn2 = (const float*)d_in[9];
  const float* bn2 = (const float*)d_in[10];
  float* out = (float*)d_out;

  float* m_sum = (float*)d_ws;                       // [NN*DF] f32
  float* m_max = m_sum + (size_t)NN * DF;            // [NN*DF] f32 (bits init -inf)

  const int total = NN * DF;
  init_kernel<<<(total + 255) / 256, 256, 0, stream>>>(m_sum, (unsigned int*)m_max, total);

  const int edgeBlocks = (NE + EW * WPB - 1) / (EW * WPB);
  edge_kernel<<<edgeBlocks, 128, 0, stream>>>(x, start, end, We1, be1, We2, be2,
                                              m_sum, m_max);

  const int nodeBlocks = (NN + NPW * WPB - 1) / (NPW * WPB);
  node_kernel<<<nodeBlocks, 128, 0, stream>>>(x, m_sum, m_max, Wn1, bn1, Wn2, bn2, out);
}